// MLP_KNN_29892972380787
// MI455X (gfx1250) — compile-verified
//
#include <hip/hip_runtime.h>
#include <math.h>

#define BB   8
#define NN   512
#define FF   16
#define EE   64
#define H1   128          // 2E
#define KK   16
#define BN   (BB*NN)      // 4096
#define BNK  (BN*KK)      // 65536

typedef __attribute__((ext_vector_type(2))) float v2f;
typedef __attribute__((ext_vector_type(8))) float v8f;

__device__ __forceinline__ unsigned long long umin64(unsigned long long a,
                                                     unsigned long long b) {
  return a < b ? a : b;
}

// ---------------------------------------------------------------------------
// Kernel 1: per-point MLP (16 -> 128 tanh -> 64 tanh), emit h and ||h||^2.
// Weights staged in LDS; one thread per point; second layer streamed so the
// register footprint stays ~90 VGPRs.
// ---------------------------------------------------------------------------
__global__ __launch_bounds__(256) void mlp_kernel(
    const float* __restrict__ X, const float* __restrict__ mask,
    const float* __restrict__ W1, const float* __restrict__ b1,
    const float* __restrict__ W2, const float* __restrict__ b2,
    float* __restrict__ h, float* __restrict__ n2) {
  __shared__ float sW1[FF * H1];   //  8 KB
  __shared__ float sW2[H1 * EE];   // 32 KB
  __shared__ float sb1[H1];
  __shared__ float sb2[EE];

  for (int i = threadIdx.x; i < FF * H1; i += 256) sW1[i] = W1[i];
  for (int i = threadIdx.x; i < H1 * EE; i += 256) sW2[i] = W2[i];
  if (threadIdx.x < H1) sb1[threadIdx.x] = b1[threadIdx.x];
  if (threadIdx.x < EE) sb2[threadIdx.x] = b2[threadIdx.x];
  __syncthreads();

  const int p = blockIdx.x * 256 + threadIdx.x;   // global point id
  const float m = mask[p];

  float x[FF];
#pragma unroll
  for (int f = 0; f < FF; ++f) x[f] = X[p * FF + f];

  float acc[EE];
#pragma unroll
  for (int e = 0; e < EE; ++e) acc[e] = sb2[e];

  for (int c = 0; c < H1; ++c) {
    float t = sb1[c];
#pragma unroll
    for (int f = 0; f < FF; ++f) t = fmaf(x[f], sW1[f * H1 + c], t);
    t = tanhf(t * m) * m;                 // layer-1 activation (masked)
#pragma unroll
    for (int e = 0; e < EE; ++e) acc[e] = fmaf(t, sW2[c * EE + e], acc[e]);
  }

  float nrm = 0.0f;
#pragma unroll
  for (int e = 0; e < EE; ++e) {
    const float v = tanhf(acc[e] * m) * m;  // layer-2 activation (masked)
    h[p * EE + e] = v;
    nrm = fmaf(v, v, nrm);
  }
  n2[p] = nrm;
}

// ---------------------------------------------------------------------------
// Kernel 2: per (batch, 16-row stripe): WMMA f32 Gram -> distance stripe in
// LDS, then stable top-16 selection per row (u64-key argmin, wave32 shuffles).
// Block = 256 threads = 8 waves. Grid = (N/16, B).
// ---------------------------------------------------------------------------
__global__ __launch_bounds__(256) void knn_kernel(
    const float* __restrict__ h, const float* __restrict__ n2,
    int* __restrict__ out) {
  __shared__ float sD[16 * NN];               // 32 KB distance stripe

  const int bt   = blockIdx.y;                // batch
  const int i0   = blockIdx.x * 16;           // stripe base row
  const int lane = threadIdx.x & 31;
  const int wave = threadIdx.x >> 5;

  const float* hb = h + bt * (NN * EE);
  const float* nb = n2 + bt * NN;

  // --- Phase A: WMMA distance tiles -------------------------------------
  // f32 16x16x4 fragment layout: lanes 0-15 hold K=0/1 (VGPR0/1),
  // lanes 16-31 hold K=2/3  ->  one aligned float2 load per fragment.
  const int arow = i0 + (lane & 15);
  const int ksub = (lane >> 4) * 2;

  v2f afrag[16];                              // A reused across 4 j-tiles
#pragma unroll
  for (int s = 0; s < 16; ++s)
    afrag[s] = *(const v2f*)(hb + arow * EE + 4 * s + ksub);

  float nrow[8];                              // ||h_i||^2 for this lane's C rows
#pragma unroll
  for (int v = 0; v < 8; ++v)
    nrow[v] = nb[i0 + v + ((lane >> 4) << 3)];

  for (int jt = wave * 4; jt < wave * 4 + 4; ++jt) {
    const int j0 = jt * 16;
    const int bcol = j0 + (lane & 15);
    v8f acc = {};
#pragma unroll
    for (int s = 0; s < 16; ++s) {
      v2f bfrag = *(const v2f*)(hb + bcol * EE + 4 * s + ksub);
      acc = __builtin_amdgcn_wmma_f32_16x16x4_f32(
          false, afrag[s], false, bfrag, (short)0, acc, false, false);
    }
    const int jj = bcol;                      // C-layout N index == lane&15
    const float nj = nb[jj];
#pragma unroll
    for (int v = 0; v < 8; ++v) {
      const int r = v + ((lane >> 4) << 3);   // row within stripe, 0..15
      const int ii = i0 + r;
      float dd = nrow[v] + nj - 2.0f * acc[v];
      if (ii == jj) dd = -1.0f;               // self always ranks first
      sD[r * NN + jj] = dd;
    }
  }
  __syncthreads();

  // --- Phase B: stable top-K per row (wave handles 2 rows) --------------
  for (int rr = 0; rr < 2; ++rr) {
    const int r = wave * 2 + rr;
    unsigned long long key[16];
#pragma unroll
    for (int t = 0; t < 16; ++t) {
      const int c = lane + 32 * t;
      unsigned u = __float_as_uint(sD[r * NN + c]);
      u = (u & 0x80000000u) ? ~u : (u | 0x80000000u);  // order-preserving map
      key[t] = ((unsigned long long)u << 32) | (unsigned)c;
    }
    const int p = bt * NN + i0 + r;
    for (int k = 0; k < KK; ++k) {
      unsigned long long best = key[0];
#pragma unroll
      for (int t = 1; t < 16; ++t) best = umin64(best, key[t]);
#pragma unroll
      for (int off = 16; off >= 1; off >>= 1) {
        const unsigned lo = __shfl_xor((unsigned)best, off, 32);
        const unsigned hi = __shfl_xor((unsigned)(best >> 32), off, 32);
        best = umin64(best, ((unsigned long long)hi << 32) | lo);
      }
      const int j = (int)(best & 0xffffffffu);
      if ((j & 31) == lane) key[j >> 5] = ~0ull;       // remove winner
      if (lane == 0) {
        out[p * KK + k]       = p;                     // track_index
        out[BNK + p * KK + k] = bt * NN + j;           // epb
      }
    }
  }
}

// ---------------------------------------------------------------------------
extern "C" void kernel_launch(void* const* d_in, const int* in_sizes, int n_in,
                              void* d_out, int out_size, void* d_ws,
                              size_t ws_size, hipStream_t stream) {
  const float* X    = (const float*)d_in[0];
  const float* mask = (const float*)d_in[1];
  const float* W1   = (const float*)d_in[2];
  const float* b1   = (const float*)d_in[3];
  const float* W2   = (const float*)d_in[4];
  const float* b2   = (const float*)d_in[5];

  float* h  = (float*)d_ws;                 // 4096*64 f32 = 1 MB
  float* n2 = h + (size_t)BN * EE;          // 4096 f32
  int*   out = (int*)d_out;                 // 2*65536 int32 edges

  mlp_kernel<<<BN / 256, 256, 0, stream>>>(X, mask, W1, b1, W2, b2, h, n2);
  knn_kernel<<<dim3(NN / 16, BB), 256, 0, stream>>>(h, n2, out);
}